// GIN_43164421324859
// MI455X (gfx1250) — compile-verified
//
#include <hip/hip_runtime.h>

#define N_NODES  100000
#define N_EDGES  1600000
#define D        128
#define N_LAYERS 4
#define N_GRAPHS 64
#define N_CLASS  10
#define BN_EPS   1e-5f
#define STAT_BLOCKS 200

typedef float v2f __attribute__((ext_vector_type(2)));
typedef float v8f __attribute__((ext_vector_type(8)));

// ---------------- column stats (mean/var), deterministic two-stage ----------
__global__ void stats_partial_kernel(const float* __restrict__ in,
                                     float* __restrict__ part, int nelem) {
    int t = threadIdx.x;
    int i0 = blockIdx.x * 256 + t;
    int stride = gridDim.x * 256;          // multiple of 128 -> feature fixed per thread
    float s = 0.f, ss = 0.f;
    for (int i = i0; i < nelem; i += stride) {
        float v = in[i];
        s += v; ss += v * v;
    }
    __shared__ float sh[512];
    sh[t] = s; sh[256 + t] = ss;
    __syncthreads();
    if (t < 128) {
        float st  = sh[t] + sh[t + 128];
        float sst = sh[256 + t] + sh[256 + t + 128];
        part[(blockIdx.x * 2 + 0) * 128 + t] = st;
        part[(blockIdx.x * 2 + 1) * 128 + t] = sst;
    }
}

__global__ void stats_finalize_kernel(const float* __restrict__ part,
                                      float* __restrict__ stats,
                                      int nblocks, float inv_n) {
    int t = threadIdx.x;   // 128 threads
    float s = 0.f, ss = 0.f;
    for (int p = 0; p < nblocks; p++) {
        s  += part[(p * 2 + 0) * 128 + t];
        ss += part[(p * 2 + 1) * 128 + t];
    }
    float mean = s * inv_n;
    float var  = ss * inv_n - mean * mean;
    stats[t] = mean;
    stats[128 + t] = var;
}

// ------------- elementwise (x-mean)*rsqrt(var+e)*gamma+beta [, relu] --------
__global__ void affine_kernel(const float* __restrict__ in, float* __restrict__ out,
                              const float* __restrict__ stats,
                              const float* __restrict__ gamma,
                              const float* __restrict__ beta,
                              float e, int do_relu, int n4) {
    int idx = blockIdx.x * blockDim.x + threadIdx.x;
    int stride = gridDim.x * blockDim.x;
    for (int i = idx; i < n4; i += stride) {
        int f = (i * 4) & 127;
        float4 v  = ((const float4*)in)[i];
        float4 m  = *(const float4*)(stats + f);
        float4 va = *(const float4*)(stats + 128 + f);
        float4 g  = gamma ? *(const float4*)(gamma + f) : make_float4(1.f, 1.f, 1.f, 1.f);
        float4 b  = beta  ? *(const float4*)(beta  + f) : make_float4(0.f, 0.f, 0.f, 0.f);
        float4 r;
        r.x = (v.x - m.x) * rsqrtf(va.x + e) * g.x + b.x;
        r.y = (v.y - m.y) * rsqrtf(va.y + e) * g.y + b.y;
        r.z = (v.z - m.z) * rsqrtf(va.z + e) * g.z + b.z;
        r.w = (v.w - m.w) * rsqrtf(va.w + e) * g.w + b.w;
        if (do_relu) {
            r.x = fmaxf(r.x, 0.f); r.y = fmaxf(r.y, 0.f);
            r.z = fmaxf(r.z, 0.f); r.w = fmaxf(r.w, 0.f);
        }
        ((float4*)out)[i] = r;
    }
}

// ---------------- z = (1+eps)*h  (fused self-term of GIN agg) ---------------
__global__ void agg_init_kernel(const float* __restrict__ h, float* __restrict__ z,
                                const float* __restrict__ eps_ptr, int n4) {
    float s = 1.0f + *eps_ptr;
    int idx = blockIdx.x * blockDim.x + threadIdx.x;
    int stride = gridDim.x * blockDim.x;
    for (int i = idx; i < n4; i += stride) {
        float4 v = ((const float4*)h)[i];
        v.x *= s; v.y *= s; v.z *= s; v.w *= s;
        ((float4*)z)[i] = v;
    }
}

// ---------------- edge scatter-add: z[dst] += h[src] ------------------------
__global__ void edge_scatter_kernel(const int* __restrict__ src,
                                    const int* __restrict__ dst,
                                    const float* __restrict__ h,
                                    float* __restrict__ z) {
    int idx = blockIdx.x * 256 + threadIdx.x;   // exactly N_EDGES*32 threads
    int e = idx >> 5;
    int q = (idx & 31) * 4;
    int s = src[e], d = dst[e];
    float4 v = *(const float4*)(h + s * 128 + q);
    float* zp = z + d * 128 + q;
    atomicAdd(zp + 0, v.x);
    atomicAdd(zp + 1, v.y);
    atomicAdd(zp + 2, v.z);
    atomicAdd(zp + 3, v.w);
}

// --------- GEMM: Cout = relu(A(N x 128) @ W(128 x 128) + bias) --------------
// Exact fp32 via V_WMMA_F32_16X16X4_F32. One wave = 16x128 output row-block.
// W staged in LDS as k-pairs so each B fragment is one ds_load_b64.
__global__ __launch_bounds__(256)
void gemm_relu_kernel(const float* __restrict__ A, const float* __restrict__ W,
                      const float* __restrict__ bias, float* __restrict__ Cout,
                      int nrows) {
    __shared__ float wl[D * D];   // 64 KB, layout: ((k>>1)*128 + col)*2 + (k&1)
    int t = threadIdx.x;
    for (int idx = t; idx < (D * D) / 4; idx += 256) {
        int gi = idx * 4;
        int k = gi >> 7;
        int c = gi & 127;
        float4 w = *(const float4*)(W + gi);
        int base = (((k >> 1) * 128) + c) * 2 + (k & 1);
        wl[base + 0] = w.x;
        wl[base + 2] = w.y;
        wl[base + 4] = w.z;
        wl[base + 6] = w.w;
    }
    __syncthreads();

    int wave = t >> 5;
    int lane = t & 31;
    int tile = blockIdx.x * 8 + wave;
    if (tile * 16 >= nrows) return;        // wave-uniform, EXEC stays all-ones

    int row0 = tile * 16;
    int lrow = lane & 15;
    int koff = (lane >> 4) << 1;           // lanes 0-15: K={0,1}; 16-31: K={2,3}
    const float* arow = A + (row0 + lrow) * 128 + koff;

    v8f acc[8] = {};
    for (int k0 = 0; k0 < 128; k0 += 4) {
        v2f a = *(const v2f*)(arow + k0);
        int pb = ((k0 + koff) >> 1) * 128 + lrow;
#pragma unroll
        for (int nt = 0; nt < 8; nt++) {
            v2f b = *(const v2f*)(wl + (pb + nt * 16) * 2);
            acc[nt] = __builtin_amdgcn_wmma_f32_16x16x4_f32(
                false, a, false, b, (short)0, acc[nt], false, false);
        }
    }

    int rowoff = (lane >> 4) * 8;          // lanes 0-15 -> rows 0-7; 16-31 -> 8-15
#pragma unroll
    for (int nt = 0; nt < 8; nt++) {
        int col = nt * 16 + lrow;
        float bv = bias[col];
        float* cp = Cout + (row0 + rowoff) * 128 + col;
#pragma unroll
        for (int r = 0; r < 8; r++) {
            float v = acc[nt][r] + bv;
            cp[r * 128] = fmaxf(v, 0.f);
        }
    }
}

// ---------------- global_add_pool with LDS staging --------------------------
__global__ void pool_kernel(const float* __restrict__ h, const int* __restrict__ batch,
                            float* __restrict__ gpool) {
    __shared__ float acc[N_GRAPHS * 128];
    int t = threadIdx.x;
    for (int i = t; i < N_GRAPHS * 128; i += 256) acc[i] = 0.f;
    __syncthreads();
    int n0 = blockIdx.x * 800;
    int f = t & 127;
    int sub = t >> 7;
    for (int it = 0; it < 400; it++) {
        int n = n0 + it * 2 + sub;
        if (n < N_NODES) {
            int g = batch[n];
            atomicAdd(&acc[g * 128 + f], h[n * 128 + f]);
        }
    }
    __syncthreads();
    for (int i = t; i < N_GRAPHS * 128; i += 256) {
        float v = acc[i];
        if (v != 0.f) atomicAdd(&gpool[i], v);
    }
}

// ---------------- tiny 3-layer MLP head (64 rows) ---------------------------
__global__ void head_kernel(const float* __restrict__ gpool,
                            const float* __restrict__ fcw1, const float* __restrict__ fcb1,
                            const float* __restrict__ fcw2, const float* __restrict__ fcb2,
                            const float* __restrict__ fcw3, const float* __restrict__ fcb3,
                            float* __restrict__ out) {
    __shared__ float g1[N_GRAPHS * 128];
    __shared__ float g2[N_GRAPHS * 64];
    int t = threadIdx.x;
    for (int o = t; o < N_GRAPHS * 128; o += 256) {
        int i = o >> 7, j = o & 127;
        float a = fcb1[j];
        for (int k = 0; k < 128; k++) a += gpool[i * 128 + k] * fcw1[k * 128 + j];
        g1[o] = fmaxf(a, 0.f);
    }
    __syncthreads();
    for (int o = t; o < N_GRAPHS * 64; o += 256) {
        int i = o >> 6, j = o & 63;
        float a = fcb2[j];
        for (int k = 0; k < 128; k++) a += g1[i * 128 + k] * fcw2[k * 64 + j];
        g2[o] = fmaxf(a, 0.f);
    }
    __syncthreads();
    for (int o = t; o < N_GRAPHS * N_CLASS; o += 256) {
        int i = o / N_CLASS, j = o % N_CLASS;
        float a = fcb3[j];
        for (int k = 0; k < 64; k++) a += g2[i * 64 + k] * fcw3[k * N_CLASS + j];
        out[o] = a;
    }
}

extern "C" void kernel_launch(void* const* d_in, const int* in_sizes, int n_in,
                              void* d_out, int out_size, void* d_ws, size_t ws_size,
                              hipStream_t stream) {
    (void)in_sizes; (void)n_in; (void)out_size; (void)ws_size;
    const float* x     = (const float*)d_in[0];
    const int*   ei    = (const int*)d_in[1];
    const int*   batch = (const int*)d_in[2];
    const float* W1    = (const float*)d_in[3];
    const float* b1    = (const float*)d_in[4];
    const float* W2    = (const float*)d_in[5];
    const float* b2    = (const float*)d_in[6];
    const float* gamma = (const float*)d_in[7];
    const float* beta  = (const float*)d_in[8];
    const float* eps   = (const float*)d_in[9];
    const float* fcw1  = (const float*)d_in[10];
    const float* fcb1  = (const float*)d_in[11];
    const float* fcw2  = (const float*)d_in[12];
    const float* fcb2  = (const float*)d_in[13];
    const float* fcw3  = (const float*)d_in[14];
    const float* fcb3  = (const float*)d_in[15];
    float* out = (float*)d_out;

    const size_t ND = (size_t)N_NODES * D;
    float* bufA  = (float*)d_ws;                      // h
    float* bufB  = bufA + ND;                         // agg / z
    float* bufC  = bufB + ND;                         // z1
    float* part  = bufC + ND;                         // STAT_BLOCKS*2*128
    float* stats = part + STAT_BLOCKS * 2 * 128;      // mean,var
    float* gpool = stats + 2 * 128;                   // 64*128

    const int n4 = (int)(ND / 4);
    const float invN = 1.0f / (float)N_NODES;
    const int* src = ei;
    const int* dst = ei + N_EDGES;
    const int edge_blocks = (N_EDGES * 32) / 256;     // exact
    const int gemm_grid = (N_NODES / 16 + 7) / 8;

    // StandardScaler: h = (x - mu) / sqrt(var)
    stats_partial_kernel<<<STAT_BLOCKS, 256, 0, stream>>>(x, part, (int)ND);
    stats_finalize_kernel<<<1, 128, 0, stream>>>(part, stats, STAT_BLOCKS, invN);
    affine_kernel<<<512, 256, 0, stream>>>(x, bufA, stats, nullptr, nullptr, 0.f, 0, n4);

    for (int l = 0; l < N_LAYERS; l++) {
        agg_init_kernel<<<512, 256, 0, stream>>>(bufA, bufB, eps + l, n4);
        edge_scatter_kernel<<<edge_blocks, 256, 0, stream>>>(src, dst, bufA, bufB);
        gemm_relu_kernel<<<gemm_grid, 256, 0, stream>>>(bufB, W1 + (size_t)l * D * D,
                                                        b1 + l * D, bufC, N_NODES);
        gemm_relu_kernel<<<gemm_grid, 256, 0, stream>>>(bufC, W2 + (size_t)l * D * D,
                                                        b2 + l * D, bufA, N_NODES);
        stats_partial_kernel<<<STAT_BLOCKS, 256, 0, stream>>>(bufA, part, (int)ND);
        stats_finalize_kernel<<<1, 128, 0, stream>>>(part, stats, STAT_BLOCKS, invN);
        affine_kernel<<<512, 256, 0, stream>>>(bufA, bufA, stats,
                                               gamma + l * D, beta + l * D, BN_EPS, 1, n4);
    }

    hipMemsetAsync(gpool, 0, (size_t)N_GRAPHS * D * sizeof(float), stream);
    pool_kernel<<<125, 256, 0, stream>>>(bufA, batch, gpool);
    head_kernel<<<1, 256, 0, stream>>>(gpool, fcw1, fcb1, fcw2, fcb2, fcw3, fcb3, out);
}